// Upsample_910533066891
// MI455X (gfx1250) — compile-verified
//
#include <hip/hip_runtime.h>
#include <hip/hip_bf16.h>
#include <stdint.h>

// Upsample-by-2 separable FIR (StyleGAN2 upfirdn2d, k=[1,3,3,1]).
// Memory-bound: ~0.64 GB traffic -> ~28us floor at 23.3 TB/s.
// CDNA5 features used: global_load_async_to_lds_b128 (ASYNCcnt) for tile
// staging, s_wait_asynccnt, non-temporal streaming stores.

#define IN_H 128
#define IN_W 128
#define RS    32            // input rows per block strip
#define LROWS (RS + 2)      // 34 rows incl. top/bottom halo
#define LSTR  132           // LDS row stride in floats (528B, 16B aligned)

__device__ __forceinline__ float4 hblend(const float* __restrict__ r, int j0, int jm1,
                                         float edge, float wE0, float wE1,
                                         float wO0, float wO1) {
    float a = r[jm1] * edge;      // in[2g-1] (0 at left image edge)
    float b = r[j0];              // in[2g]
    float c = r[j0 + 1];          // in[2g+1]
    float d = r[j0 + 2];          // in[2g+2] (col 128 is a zeroed halo)
    float4 h;
    h.x = wE0 * a + wE1 * b;      // out col 4g   (even, j=2g)
    h.y = wO0 * b + wO1 * c;      // out col 4g+1 (odd,  j=2g)
    h.z = wE0 * b + wE1 * c;      // out col 4g+2 (even, j=2g+1)
    h.w = wO0 * c + wO1 * d;      // out col 4g+3 (odd,  j=2g+1)
    return h;
}

__global__ __launch_bounds__(256)
void upsample2x_fir(const float* __restrict__ x, const float* __restrict__ k1d,
                    float* __restrict__ out, int ohw) {
    __shared__ float lds[LROWS * LSTR];

    const int tid   = threadIdx.x;
    const int plane = blockIdx.x >> 2;       // n*C + c
    const int strip = blockIdx.x & 3;
    const int r0    = strip * RS;

    const float* xp = x + (size_t)plane * (IN_H * IN_W);
    float*       op = out + (size_t)plane * ((size_t)ohw * (size_t)ohw);

    // Separable 1D weights: g = k1d * 2/sum(k1d) -> [.25,.75,.75,.25]
    const float k0 = k1d[0], k1 = k1d[1], k2 = k1d[2], k3 = k1d[3];
    const float s  = 2.0f / (k0 + k1 + k2 + k3);
    const float wE0 = k3 * s, wE1 = k1 * s;   // even phase: x[m-1], x[m]
    const float wO0 = k2 * s, wO1 = k0 * s;   // odd phase:  x[m],   x[m+1]

    // ---- Stage rows r0-1 .. r0+32 into LDS via async global->LDS (b128) ----
    for (int t = tid; t < LROWS * (IN_W / 4); t += 256) {
        const int row = t >> 5;               // 0..33
        const int col = (t & 31) << 2;        // 0,4,...,124
        const int gr  = r0 - 1 + row;         // global input row (-1..128)
        const uint32_t laddr = (uint32_t)(uintptr_t)&lds[row * LSTR + col];
        if ((unsigned)gr < (unsigned)IN_H) {
            const uint64_t gaddr =
                (uint64_t)(uintptr_t)(xp + (size_t)gr * IN_W + col);
            asm volatile("global_load_async_to_lds_b128 %0, %1, off"
                         :: "v"(laddr), "v"(gaddr) : "memory");
        } else {
            *(float4*)&lds[row * LSTR + col] = make_float4(0.f, 0.f, 0.f, 0.f);
        }
    }
    if (tid < LROWS) lds[tid * LSTR + IN_W] = 0.0f;   // zero halo column 128

    asm volatile("s_wait_asynccnt 0" ::: "memory");
    __syncthreads();

    // ---- Compute: thread -> column group g (out cols 4g..4g+3), row slot rp ----
    const int   g    = tid & 63;
    const int   rp   = tid >> 6;              // 0..3
    const int   j0   = g << 1;                // input col 2g
    const int   jm1  = (g > 0) ? (j0 - 1) : 0;
    const float edge = (g > 0) ? 1.0f : 0.0f;
    const bool  skipLastCol = (g == 63) && (ohw == 255);

#pragma unroll
    for (int k = 0; k < 8; ++k) {
        const int li = rp + (k << 2) + 1;     // LDS row of input row i (1..32)
        const float* rA = &lds[(li - 1) * LSTR];
        const float* rB = &lds[ li      * LSTR];
        const float* rC = &lds[(li + 1) * LSTR];

        const float4 hA = hblend(rA, j0, jm1, edge, wE0, wE1, wO0, wO1);
        const float4 hB = hblend(rB, j0, jm1, edge, wE0, wE1, wO0, wO1);
        const float4 hC = hblend(rC, j0, jm1, edge, wE0, wE1, wO0, wO1);

        float4 e, o;                          // even row 2i, odd row 2i+1
        e.x = wE0 * hA.x + wE1 * hB.x;  o.x = wO0 * hB.x + wO1 * hC.x;
        e.y = wE0 * hA.y + wE1 * hB.y;  o.y = wO0 * hB.y + wO1 * hC.y;
        e.z = wE0 * hA.z + wE1 * hB.z;  o.z = wO0 * hB.z + wO1 * hC.z;
        e.w = wE0 * hA.w + wE1 * hB.w;  o.w = wO0 * hB.w + wO1 * hC.w;

        const int i  = r0 + li - 1;           // input row index
        const int oy = i << 1;                // even output row (<= 254)
        float* pe = op + (size_t)oy * (size_t)ohw + (j0 << 1);
        __builtin_nontemporal_store(e.x, pe + 0);
        __builtin_nontemporal_store(e.y, pe + 1);
        __builtin_nontemporal_store(e.z, pe + 2);
        if (!skipLastCol) __builtin_nontemporal_store(e.w, pe + 3);
        if (oy + 1 < ohw) {
            float* po = pe + ohw;
            __builtin_nontemporal_store(o.x, po + 0);
            __builtin_nontemporal_store(o.y, po + 1);
            __builtin_nontemporal_store(o.z, po + 2);
            if (!skipLastCol) __builtin_nontemporal_store(o.w, po + 3);
        }
    }
}

extern "C" void kernel_launch(void* const* d_in, const int* in_sizes, int n_in,
                              void* d_out, int out_size, void* d_ws, size_t ws_size,
                              hipStream_t stream) {
    (void)n_in; (void)d_ws; (void)ws_size;
    const float* x   = (const float*)d_in[0];
    const float* k1d = (const float*)d_in[1];
    float*       out = (float*)d_out;

    const int planes = in_sizes[0] / (IN_H * IN_W);          // 16*128 = 2048
    const int perPlane = out_size / planes;                  // 65025 or 65536
    const int ohw = (perPlane == 256 * 256) ? 256 : 255;     // XLA lhs_dilation -> 255

    dim3 grid(planes * 4), block(256);
    upsample2x_fir<<<grid, block, 0, stream>>>(x, k1d, out, ohw);
}